// TemporalCausalGraph_17033840296043
// MI455X (gfx1250) — compile-verified
//
#include <hip/hip_runtime.h>
#include <hip/hip_bf16.h>

// ---------------------------------------------------------------------------
// Model dims
// ---------------------------------------------------------------------------
#define NND   512   // N_NODES
#define HID   64    // HIDDEN
#define LAT   256   // LATENT
#define TST   10    // T_STEPS
#define BSZ   64    // BATCH
#define NHD   4     // N_HEADS
#define NLY   2     // N_LAYERS
#define NNSQ  (NND*NND)

typedef __attribute__((ext_vector_type(2))) float v2f;
typedef __attribute__((ext_vector_type(8))) float v8f;

// ---------------------------------------------------------------------------
// Generic strided GEMM: C[M,N] = act(A[M,K] @ B[K,N] + bias)
// One wave computes a 16x64 strip (4 x 16x16 tiles) with V_WMMA_F32_16X16X4_F32.
// The A fragment is loaded once per K-step and reused by 4 back-to-back WMMAs.
// Requires: M%64==0 (4 waves/block along M), N%64==0, K%4==0. EXEC all-ones.
//
// Fragment layout per CDNA5 ISA 7.12.2:
//   A (16x4 f32, v2f): lanes 0-15 -> M=lane, K=kk+{0,1}; lanes 16-31 -> K=kk+{2,3}
//   B (4x16 f32, v2f): VGPR v -> row kk+half*2+v, col = n0 + (lane&15)
//   C (16x16 f32, v8f): VGPR v -> row m0 + half*8 + v, col = n0 + (lane&15)
// act: 0 = none, 1 = relu, 2 = sigmoid
// ---------------------------------------------------------------------------
__global__ void wmma_gemm_f32(const float* __restrict__ A, int lda,
                              const float* __restrict__ B, int ldb,
                              float* __restrict__ C, int ldc,
                              const float* __restrict__ bias,
                              int K, int act) {
  const int wave = threadIdx.x >> 5;
  const int lane = threadIdx.x & 31;
  const int half = lane >> 4;
  const int l16  = lane & 15;
  const int n0 = blockIdx.x * 64;                 // 64-column strip per wave
  const int m0 = (blockIdx.y * 4 + wave) * 16;    // 16-row tile per wave

  const float* Ap = A + (size_t)(m0 + l16) * lda + half * 2;
  const float* Bp = B + (size_t)(half * 2) * ldb + n0 + l16;

  v8f acc0 = {}, acc1 = {}, acc2 = {}, acc3 = {};
  for (int kk = 0; kk < K; kk += 4) {
    __builtin_prefetch(&Ap[kk + 16], 0, 1);       // global_prefetch_b8
    v2f a;
    a.x = Ap[kk];
    a.y = Ap[kk + 1];
    const float* Br0 = Bp + (size_t)kk * ldb;
    const float* Br1 = Br0 + ldb;
    v2f b0, b1, b2, b3;
    b0.x = Br0[0];  b0.y = Br1[0];
    b1.x = Br0[16]; b1.y = Br1[16];
    b2.x = Br0[32]; b2.y = Br1[32];
    b3.x = Br0[48]; b3.y = Br1[48];
    acc0 = __builtin_amdgcn_wmma_f32_16x16x4_f32(false, a, false, b0, (short)0, acc0, false, false);
    acc1 = __builtin_amdgcn_wmma_f32_16x16x4_f32(false, a, false, b1, (short)0, acc1, false, false);
    acc2 = __builtin_amdgcn_wmma_f32_16x16x4_f32(false, a, false, b2, (short)0, acc2, false, false);
    acc3 = __builtin_amdgcn_wmma_f32_16x16x4_f32(false, a, false, b3, (short)0, acc3, false, false);
  }

  float* Cp = C + (size_t)(m0 + half * 8) * ldc + n0 + l16;
#define WMMA_EPI(ACC, OFF)                                        \
  {                                                               \
    float bv = bias ? bias[n0 + (OFF) + l16] : 0.0f;              \
    _Pragma("unroll")                                             \
    for (int v = 0; v < 8; ++v) {                                 \
      float val = ACC[v] + bv;                                    \
      if (act == 1) val = fmaxf(val, 0.0f);                       \
      else if (act == 2) val = 1.0f / (1.0f + expf(-val));        \
      Cp[(size_t)v * ldc + (OFF)] = val;                          \
    }                                                             \
  }
  WMMA_EPI(acc0, 0)
  WMMA_EPI(acc1, 16)
  WMMA_EPI(acc2, 32)
  WMMA_EPI(acc3, 48)
#undef WMMA_EPI
}

// ---------------------------------------------------------------------------
// Elementwise / small kernels
// ---------------------------------------------------------------------------
__device__ __forceinline__ unsigned hashu(unsigned x, unsigned seed) {
  x ^= seed; x *= 0x85ebca6bu; x ^= x >> 13; x *= 0xc2b2ae35u; x ^= x >> 16;
  return x;
}
__device__ __forceinline__ float sigf(float x) { return 1.0f / (1.0f + expf(-x)); }

// [B*T, H] (row b*T+t) -> [T*B, H] (row t*B+b)
__global__ void permute_bt(const float* __restrict__ in, float* __restrict__ out) {
  int idx = blockIdx.x * blockDim.x + threadIdx.x;
  if (idx >= BSZ * TST * HID) return;
  int c = idx % HID, row = idx / HID;
  int b = row / TST, t = row % TST;
  out[(t * BSZ + b) * HID + c] = in[idx];
}

// tiny multi-head attention: S=10, hd=16; one block per (b,h)
__global__ void attn_kernel(const float* __restrict__ qkv, float* __restrict__ o) {
  const int b = blockIdx.x >> 2, h = blockIdx.x & 3;
  const int tid = threadIdx.x;
  __shared__ float q[TST][16], k[TST][16], v[TST][16], att[TST][TST];
  for (int i = tid; i < TST * 16; i += 64) {
    int s = i >> 4, d = i & 15;
    int base = (s * BSZ + b) * (3 * HID) + h * 16 + d;
    q[s][d] = qkv[base];
    k[s][d] = qkv[base + HID];
    v[s][d] = qkv[base + 2 * HID];
  }
  __syncthreads();
  for (int i = tid; i < TST * TST; i += 64) {
    int s = i / TST, t = i % TST;
    float acc = 0.f;
    for (int d = 0; d < 16; ++d) acc += q[s][d] * k[t][d];
    att[s][t] = acc * 0.25f;  // 1/sqrt(16)
  }
  __syncthreads();
  if (tid < TST) {
    float m = -1e30f;
    for (int t = 0; t < TST; ++t) m = fmaxf(m, att[tid][t]);
    float sum = 0.f;
    for (int t = 0; t < TST; ++t) { float e = expf(att[tid][t] - m); att[tid][t] = e; sum += e; }
    float inv = 1.0f / sum;
    for (int t = 0; t < TST; ++t) att[tid][t] *= inv;
  }
  __syncthreads();
  for (int i = tid; i < TST * 16; i += 64) {
    int s = i >> 4, d = i & 15;
    float acc = 0.f;
    for (int t = 0; t < TST; ++t) acc += att[s][t] * v[t][d];
    o[(s * BSZ + b) * HID + h * 16 + d] = acc;
  }
}

// x = LayerNorm(x + o) * g + b ; one block (64 thr) per row, D=64
__global__ void ln_residual(float* __restrict__ x, const float* __restrict__ o,
                            const float* __restrict__ g, const float* __restrict__ b) {
  const int row = blockIdx.x, t = threadIdx.x;
  __shared__ float red[HID];
  float v = x[row * HID + t] + o[row * HID + t];
  red[t] = v; __syncthreads();
  for (int s = 32; s > 0; s >>= 1) { if (t < s) red[t] += red[t + s]; __syncthreads(); }
  float mu = red[0] / HID; __syncthreads();
  float d = v - mu;
  red[t] = d * d; __syncthreads();
  for (int s = 32; s > 0; s >>= 1) { if (t < s) red[t] += red[t + s]; __syncthreads(); }
  float rstd = rsqrtf(red[0] / HID + 1e-5f);
  x[row * HID + t] = d * rstd * g[t] + b[t];
}

// x_instT[n,b] = xw2[t=0]; lag[b,n] = mean_{t=1..9} xw2; lagT = lag^T
__global__ void inst_lag_kernel(const float* __restrict__ xw2, float* __restrict__ xinstT,
                                float* __restrict__ lag, float* __restrict__ lagT) {
  int idx = blockIdx.x * blockDim.x + threadIdx.x;
  if (idx >= BSZ * NND) return;
  int b = idx / NND, n = idx % NND;
  xinstT[n * BSZ + b] = xw2[b * NND + n];
  float acc = 0.f;
  for (int t = 1; t < TST; ++t) acc += xw2[(t * BSZ + b) * NND + n];
  acc *= (1.0f / 9.0f);
  lag[b * NND + n] = acc;
  lagT[n * BSZ + b] = acc;
}

// gumbel-softmax row + sigmoid + clip + zero-diag + replace_zero
__global__ void adj_kernel(const float* __restrict__ lg, const float* __restrict__ es,
                           const float* __restrict__ prior, float* __restrict__ out,
                           unsigned seed) {
  const int i = blockIdx.x, t = threadIdx.x;
  __shared__ float red[256];
  float m = -1e30f;
  for (int j = t; j < NND; j += 256) {
    unsigned h = hashu((unsigned)(i * NND + j), seed);
    float u = ((float)h + 0.5f) * 2.3283064e-10f;
    float g = -logf(-logf(u));
    m = fmaxf(m, lg[i * NND + j] + g);
  }
  red[t] = m; __syncthreads();
  for (int s = 128; s > 0; s >>= 1) { if (t < s) red[t] = fmaxf(red[t], red[t + s]); __syncthreads(); }
  m = red[0]; __syncthreads();
  float sum = 0.f;
  for (int j = t; j < NND; j += 256) {
    unsigned h = hashu((unsigned)(i * NND + j), seed);
    float u = ((float)h + 0.5f) * 2.3283064e-10f;
    float g = -logf(-logf(u));
    sum += expf(lg[i * NND + j] + g - m);
  }
  red[t] = sum; __syncthreads();
  for (int s = 128; s > 0; s >>= 1) { if (t < s) red[t] += red[t + s]; __syncthreads(); }
  float inv = 1.0f / red[0];
  for (int j = t; j < NND; j += 256) {
    unsigned h = hashu((unsigned)(i * NND + j), seed);
    float u = ((float)h + 0.5f) * 2.3283064e-10f;
    float g = -logf(-logf(u));
    float a = expf(lg[i * NND + j] + g - m) * inv;
    float v = sigf(es[i * NND + j] + prior[i * NND + j] + a);
    v = fminf(fmaxf(v, 0.0f), 1.0f);
    if (j == i) v = 0.0f;
    if (v == 0.0f) v = 1e-8f;
    out[i * NND + j] = v;
  }
}

// z2buf = s * Z[:,128:] + tbuf
__global__ void flow_z2(const float* __restrict__ Z, const float* __restrict__ s,
                        const float* __restrict__ tb, float* __restrict__ z2buf) {
  int idx = blockIdx.x * blockDim.x + threadIdx.x;
  if (idx >= NND * (LAT / 2)) return;
  int n = idx >> 7, c = idx & 127;
  z2buf[idx] = s[idx] * Z[n * LAT + 128 + c] + tb[idx];
}

// temb[n,t,c] = sin/cos(time_context * exp(c*10/127))
__global__ void temb_kernel(const int* __restrict__ tc, float* __restrict__ temb) {
  int idx = blockIdx.x * blockDim.x + threadIdx.x;
  if (idx >= NND * TST * LAT) return;
  int c = idx % LAT, r = idx / LAT;
  int t = r % TST, n = r / TST;
  float freq = expf((float)(c & 127) * (10.0f / 127.0f));
  float a = (float)tc[n * TST + t] * freq;
  temb[idx] = (c < 128) ? sinf(a) : cosf(a);
}

__global__ void fill_zero(float* __restrict__ p, int n) {
  int i = blockIdx.x * blockDim.x + threadIdx.x;
  if (i < n) p[i] = 0.0f;
}

// one GRU step: gates + h update + tout store
__global__ void gru_gate(const float* __restrict__ gi, const float* __restrict__ gh,
                         float* __restrict__ h, float* __restrict__ tout, int t) {
  int idx = blockIdx.x * blockDim.x + threadIdx.x;
  if (idx >= NND * 128) return;
  int n = idx >> 7, c = idx & 127;
  const float* gir = gi + (size_t)(n * TST + t) * 384;
  const float* ghr = gh + (size_t)n * 384;
  float r  = sigf(gir[c] + ghr[c]);
  float z  = sigf(gir[128 + c] + ghr[128 + c]);
  float nn = tanhf(gir[256 + c] + r * ghr[256 + c]);
  float hn = (1.0f - z) * nn + z * h[idx];
  h[idx] = hn;
  tout[(size_t)(n * TST + t) * 128 + c] = hn;
}

// zc = concat(z1, z2 + time-pool(tout))
__global__ void build_zc(const float* __restrict__ Z, const float* __restrict__ z2buf,
                         const float* __restrict__ tout, const float* __restrict__ tpw,
                         const float* __restrict__ tpb, float* __restrict__ zc) {
  int idx = blockIdx.x * blockDim.x + threadIdx.x;
  if (idx >= NND * LAT) return;
  int n = idx >> 8, c = idx & 255;
  float v;
  if (c < 128) {
    v = Z[n * LAT + c];
  } else {
    int cc = c - 128;
    float acc = z2buf[n * 128 + cc];
    for (int t = 0; t < TST; ++t) acc += tout[(size_t)(n * TST + t) * 128 + cc] * tpw[t];
    v = acc + tpb[0];
  }
  zc[idx] = v;
}

// e[i,h,j] = sum_c leaky_relu(xr[i,h,c] + xl[j,h,c], 0.2) * att[h,c]
__global__ void gat_e_kernel(const float* __restrict__ xl, const float* __restrict__ xr,
                             const float* __restrict__ att, float* __restrict__ e) {
  const int i = blockIdx.x, h = blockIdx.y, t = threadIdx.x;
  __shared__ float xrs[64], atts[64];
  if (t < 64) { xrs[t] = xr[i * 256 + h * 64 + t]; atts[t] = att[h * 64 + t]; }
  __syncthreads();
  for (int j = t; j < NND; j += 256) {
    float acc = 0.f;
    const float* xlr = xl + (size_t)j * 256 + h * 64;
    for (int c = 0; c < 64; ++c) {
      float a = xrs[c] + xlr[c];
      a = (a > 0.f) ? a : 0.2f * a;
      acc += a * atts[c];
    }
    e[((size_t)i * 4 + h) * NND + j] = acc;
  }
}

// in-place row softmax (width 512)
__global__ void softmax_rows(float* __restrict__ x) {
  const int row = blockIdx.x, t = threadIdx.x;
  __shared__ float red[256];
  float* xr = x + (size_t)row * NND;
  float m = -1e30f;
  for (int j = t; j < NND; j += 256) m = fmaxf(m, xr[j]);
  red[t] = m; __syncthreads();
  for (int s = 128; s > 0; s >>= 1) { if (t < s) red[t] = fmaxf(red[t], red[t + s]); __syncthreads(); }
  m = red[0]; __syncthreads();
  float sum = 0.f;
  for (int j = t; j < NND; j += 256) { float e = expf(xr[j] - m); xr[j] = e; sum += e; }
  red[t] = sum; __syncthreads();
  for (int s = 128; s > 0; s >>= 1) { if (t < s) red[t] += red[t + s]; __syncthreads(); }
  float inv = 1.0f / red[0];
  for (int j = t; j < NND; j += 256) xr[j] *= inv;
}

// split ml -> mean, scale = exp(0.5*clip(logvar,-5,2))
__global__ void mean_scale(const float* __restrict__ ml, float* __restrict__ out) {
  int idx = blockIdx.x * blockDim.x + threadIdx.x;
  if (idx >= NNSQ) return;
  int i = idx >> 9, j = idx & 511;
  out[2 * NNSQ + idx] = ml[(size_t)i * 1024 + j];
  float lv = ml[(size_t)i * 1024 + 512 + j];
  lv = fminf(fmaxf(lv, -5.0f), 2.0f);
  out[3 * NNSQ + idx] = expf(0.5f * lv);
}

// ---------------------------------------------------------------------------
// Host orchestration
// ---------------------------------------------------------------------------
extern "C" void kernel_launch(void* const* d_in, const int* in_sizes, int n_in,
                              void* d_out, int out_size, void* d_ws, size_t ws_size,
                              hipStream_t stream) {
  const float* X      = (const float*)d_in[0];
  const float* Z      = (const float*)d_in[1];
  const float* es_now = (const float*)d_in[2];
  const float* es_lag = (const float*)d_in[3];
  const float* prior  = (const float*)d_in[4];
  const float* wp1    = (const float*)d_in[5];
  const float* bp1    = (const float*)d_in[6];
  const float* e_inw  = (const float*)d_in[7];
  const float* e_inb  = (const float*)d_in[8];
  const float* e_ow   = (const float*)d_in[9];
  const float* e_ob   = (const float*)d_in[10];
  const float* e_l1w  = (const float*)d_in[11];
  const float* e_l1b  = (const float*)d_in[12];
  const float* e_l2w  = (const float*)d_in[13];
  const float* e_l2b  = (const float*)d_in[14];
  const float* e_g1   = (const float*)d_in[15];
  const float* e_b1   = (const float*)d_in[16];
  const float* e_g2   = (const float*)d_in[17];
  const float* e_b2   = (const float*)d_in[18];
  const float* wp2    = (const float*)d_in[19];
  const float* bp2    = (const float*)d_in[20];
  const float* fs_w   = (const float*)d_in[21];
  const float* fs_b   = (const float*)d_in[22];
  const float* ft_w   = (const float*)d_in[23];
  const float* ft_b   = (const float*)d_in[24];
  const float* g_wih  = (const float*)d_in[25];
  const float* g_whh  = (const float*)d_in[26];
  const float* g_bih  = (const float*)d_in[27];
  const float* g_bhh  = (const float*)d_in[28];
  const float* tp_w   = (const float*)d_in[29];
  const float* tp_b   = (const float*)d_in[30];
  const float* ltn_w  = (const float*)d_in[31];
  const float* ltn_b  = (const float*)d_in[32];
  const float* gat_wl = (const float*)d_in[33];
  const float* gat_bl = (const float*)d_in[34];
  const float* gat_wr = (const float*)d_in[35];
  const float* gat_br = (const float*)d_in[36];
  const float* gat_at = (const float*)d_in[37];
  const float* gat_bi = (const float*)d_in[38];
  const float* gl_w   = (const float*)d_in[39];
  const float* gl_b   = (const float*)d_in[40];
  const int*   timec  = (const int*)d_in[41];
  float* out = (float*)d_out;

  // workspace carve-out (~32MB fp32)
  float* W = (float*)d_ws;
  size_t off = 0;
  auto alloc = [&](size_t n) { float* p = W + off; off += n; return p; };
  float* xbt    = alloc(640 * 64);
  float* xT     = alloc(640 * 64);
  float* qkv    = alloc(640 * 192);
  float* obuf   = alloc(640 * 64);
  float* tmp    = alloc(640 * 64);
  float* tmp2   = alloc(640 * 64);
  float* xw2    = alloc(640 * 512);
  float* xinstT = alloc(512 * 64);
  float* lag    = alloc(64 * 512);
  float* lagT   = alloc(512 * 64);
  float* lognow = alloc(512 * 512);
  float* loglag = alloc(512 * 512);
  float* sbuf   = alloc(512 * 128);
  float* tbuf   = alloc(512 * 128);
  float* z2buf  = alloc(512 * 128);
  float* temb   = alloc(512 * 10 * 256);
  float* gi     = alloc(512 * 10 * 384);
  float* gh     = alloc(512 * 384);
  float* hbuf   = alloc(512 * 128);
  float* tout   = alloc(512 * 10 * 128);
  float* zc     = alloc(512 * 256);
  float* Zn     = alloc(512 * 512);
  float* xl     = alloc(512 * 256);
  float* xr     = alloc(512 * 256);
  float* ebuf   = alloc(512 * 4 * 512);
  float* xemb   = alloc(512 * 256);
  float* ml     = alloc(512 * 1024);

  auto gemm = [&](const float* A, int lda, const float* B, int ldb, float* C, int ldc,
                  const float* bias, int M, int N, int K, int act) {
    dim3 grid(N / 64, M / 64);
    wmma_gemm_f32<<<grid, 128, 0, stream>>>(A, lda, B, ldb, C, ldc, bias, K, act);
  };
  auto ew = [&](int n) { return dim3((n + 255) / 256); };

  // 1) input projection + seq-first permute
  gemm(X, NND, wp1, HID, xbt, HID, bp1, BSZ * TST, HID, NND, 0);
  permute_bt<<<ew(640 * 64), 256, 0, stream>>>(xbt, xT);

  // 2) transformer encoder layers (post-LN, relu FFN)
  for (int l = 0; l < NLY; ++l) {
    gemm(xT, HID, e_inw + l * HID * 3 * HID, 3 * HID, qkv, 3 * HID,
         e_inb + l * 3 * HID, 640, 3 * HID, HID, 0);
    attn_kernel<<<BSZ * NHD, 64, 0, stream>>>(qkv, obuf);
    gemm(obuf, HID, e_ow + l * HID * HID, HID, tmp, HID, e_ob + l * HID, 640, HID, HID, 0);
    ln_residual<<<640, 64, 0, stream>>>(xT, tmp, e_g1 + l * HID, e_b1 + l * HID);
    gemm(xT, HID, e_l1w + l * HID * HID, HID, tmp2, HID, e_l1b + l * HID, 640, HID, HID, 1);
    gemm(tmp2, HID, e_l2w + l * HID * HID, HID, tmp, HID, e_l2b + l * HID, 640, HID, HID, 0);
    ln_residual<<<640, 64, 0, stream>>>(xT, tmp, e_g2 + l * HID, e_b2 + l * HID);
  }

  // 3) output projection, instantaneous / lagged summaries
  gemm(xT, HID, wp2, NND, xw2, NND, bp2, 640, NND, HID, 0);
  inst_lag_kernel<<<ew(BSZ * NND), 256, 0, stream>>>(xw2, xinstT, lag, lagT);

  // 4) gram matrices (K=64) + gumbel-softmax adjacency
  gemm(xinstT, BSZ, xw2, NND, lognow, NND, nullptr, NND, NND, BSZ, 0);       // x_inst^T x_inst
  gemm(lagT, BSZ, lag, NND, loglag, NND, nullptr, NND, NND, BSZ, 2);         // sigmoid(...)
  adj_kernel<<<NND, 256, 0, stream>>>(lognow, es_now, prior, out, 0x9E3779B9u);
  adj_kernel<<<NND, 256, 0, stream>>>(loglag, es_lag, prior, out + NNSQ, 0x85EBCA77u);

  // 5) RealNVP flow on Z
  gemm(Z, LAT, fs_w, 128, sbuf, 128, fs_b, NND, 128, 128, 2);   // s = sigmoid(z1@fs+b)
  gemm(Z, LAT, ft_w, 128, tbuf, 128, ft_b, NND, 128, 128, 0);   // t = z1@ft+b
  flow_z2<<<ew(NND * 128), 256, 0, stream>>>(Z, sbuf, tbuf, z2buf);

  // 6) time embedding + GRU
  temb_kernel<<<ew(NND * TST * LAT), 256, 0, stream>>>(timec, temb);
  gemm(temb, LAT, g_wih, 384, gi, 384, g_bih, NND * TST, 384, LAT, 0);
  fill_zero<<<ew(NND * 128), 256, 0, stream>>>(hbuf, NND * 128);
  for (int t = 0; t < TST; ++t) {
    gemm(hbuf, 128, g_whh, 384, gh, 384, g_bhh, NND, 384, 128, 0);
    gru_gate<<<ew(NND * 128), 256, 0, stream>>>(gi, gh, hbuf, tout, t);
  }
  build_zc<<<ew(NND * LAT), 256, 0, stream>>>(Z, z2buf, tout, tp_w, tp_b, zc);

  // 7) latent -> node features
  gemm(zc, LAT, ltn_w, NND, Zn, NND, ltn_b, NND, NND, LAT, 0);

  // 8) GATv2
  gemm(Zn, NND, gat_wl, 256, xl, 256, gat_bl, NND, 256, NND, 0);
  gemm(Zn, NND, gat_wr, 256, xr, 256, gat_br, NND, 256, NND, 0);
  gat_e_kernel<<<dim3(NND, NHD), 256, 0, stream>>>(xl, xr, gat_at, ebuf);
  softmax_rows<<<NND * NHD, 256, 0, stream>>>(ebuf);
  for (int h = 0; h < NHD; ++h) {
    // x_emb[:, h*64:(h+1)*64] = alpha_h [512x512] @ xl_h [512x64] + bias_h
    gemm(ebuf + (size_t)h * NND, NHD * NND, xl + h * 64, 256,
         xemb + h * 64, 256, gat_bi + h * 64, NND, 64, NND, 0);
  }

  // 9) Laplace head
  gemm(xemb, 256, gl_w, 2 * NND, ml, 2 * NND, gl_b, NND, 2 * NND, 256, 0);
  mean_scale<<<ew(NNSQ), 256, 0, stream>>>(ml, out);
}